// LSTM_6597069766957
// MI455X (gfx1250) — compile-verified
//
#include <hip/hip_runtime.h>
#include <hip/hip_bf16.h>
#include <cstdint>

// LSTM LM forward on MI455X (gfx1250, wave32, WMMA).
// FP32 path via V_WMMA_F32_16X16X4_F32 to match the fp32 reference exactly.
//
// Per-wave macro-tiles of 64x16 with 4 accumulators share one fragment per
// k-step (1.25 loads/WMMA instead of 2.0) -- the fp32 WMMA pipe is the
// throughput ceiling, so the inner loop is kept load-light. All GEMM operands
// (Wh* 16MB, Why 41MB, hs 33MB) are L2-resident (192MB L2); only the 327MB
// logits stream hits HBM.
//
// d_ws layout (floats):
//   xi,xf,xc,xo : 4 x [S,B,H] gate pre-activations (input side)
//   hs          : [S,B,H] hidden states (step s writes hs[s], step s+1 reads it)
//   cbuf        : [B,H] running cell state
// Total ~168 MB.

#define VOCAB  10000
#define EMBED  512
#define HIDDEN 1024
#define BATCH  32
#define SEQ    256

typedef __attribute__((ext_vector_type(2))) float v2f;
typedef __attribute__((ext_vector_type(8))) float v8f;

__device__ __forceinline__ v8f wmma_f32(v2f a, v2f b, v8f c) {
    // D = A(16x4) * B(4x16) + C(16x16), fp32
    return __builtin_amdgcn_wmma_f32_16x16x4_f32(
        /*neg_a=*/false, a, /*neg_b=*/false, b,
        /*c_mod=*/(short)0, c, /*reuse_a=*/false, /*reuse_b=*/false);
}

__device__ __forceinline__ float sigm(float x) {
    return 1.0f / (1.0f + __expf(-x));
}
__device__ __forceinline__ float ftanh(float x) {
    float e = __expf(2.0f * x);
    return (e - 1.0f) / (e + 1.0f);
}

struct GateParams {
    const float* W[4];     // [H, E] each
    const float* bias[4];  // [H]
    float*       out[4];   // [S,B,H] each
};

// ---------------------------------------------------------------------------
// Kernel 1: input-side gate projections for all timesteps.
//   out[g][s*B + b][h] = emb[x[b,s]] . W[g][h,:] + bias[g][h]
// One wave per block; 64(M) x 16(N) macro-tile = 4 accumulators sharing one
// W fragment per k-step.  Grid = (8192/64, 1024/16, 4 gates).
// ---------------------------------------------------------------------------
__global__ __launch_bounds__(32) void gates_x_kernel(
    const int* __restrict__ x, const float* __restrict__ emb, GateParams p)
{
    const int lane = threadIdx.x & 31;
    const int mi = lane & 15;
    const int hi = lane >> 4;            // 0: K0/K1 half, 1: K2/K3 half
    const int g  = blockIdx.z;
    const int m0 = blockIdx.x * 64;      // row tile over [S,B] (row = s*32 + b)
    const int n0 = blockIdx.y * 16;

    const int n = n0 + mi;
    const float* bp = p.W[g] + (size_t)n * EMBED + 2 * hi;     // B col (W row)

    const float* ap[4];
    #pragma unroll
    for (int t = 0; t < 4; ++t) {
        const int row = m0 + 16 * t + mi;
        const int s = row >> 5;
        const int b = row & 31;
        const int tok = x[b * SEQ + s];
        ap[t] = emb + (size_t)tok * EMBED + 2 * hi;            // A rows
    }

    v8f acc0 = {}, acc1 = {}, acc2 = {}, acc3 = {};
    #pragma unroll 2
    for (int k = 0; k < EMBED; k += 4) {
        v2f bf = *(const v2f*)(bp + k);
        v2f a0 = *(const v2f*)(ap[0] + k);
        v2f a1 = *(const v2f*)(ap[1] + k);
        v2f a2 = *(const v2f*)(ap[2] + k);
        v2f a3 = *(const v2f*)(ap[3] + k);
        acc0 = wmma_f32(a0, bf, acc0);
        acc1 = wmma_f32(a1, bf, acc1);
        acc2 = wmma_f32(a2, bf, acc2);
        acc3 = wmma_f32(a3, bf, acc3);
    }

    const float bias = p.bias[g][n];
    float* outg = p.out[g];
    v8f accs[4] = {acc0, acc1, acc2, acc3};
    #pragma unroll
    for (int t = 0; t < 4; ++t) {
        #pragma unroll
        for (int j = 0; j < 8; ++j) {
            const int orow = m0 + 16 * t + j + 8 * hi;   // C: VGPR j -> M=j (+8 hi)
            outg[(size_t)orow * HIDDEN + n] = accs[t][j] + bias;
        }
    }
}

// ---------------------------------------------------------------------------
// Kernel 2: one recurrence step (launched 256x).
// Each wave computes one 16x16 (batch x hidden) tile of ALL FOUR gate GEMMs,
// sharing the h_{t-1} A-fragment across the 4 WMMAs per k-step, then applies
// the LSTM pointwise update and writes h_t into hs[s] (read as A next step).
// Wh* (16 MB total) stays resident in the 192MB L2 across all 256 steps.
// Grid = (32/16, 1024/16) = (2,64), 1 wave per block for max WGP spread.
// ---------------------------------------------------------------------------
__global__ __launch_bounds__(32) void lstm_step_kernel(
    const float* __restrict__ hprev,   // [B,H]
    float* __restrict__ hout,          // [B,H] = hs[s]
    float* __restrict__ cbuf,          // [B,H]
    const float* __restrict__ xi, const float* __restrict__ xf,
    const float* __restrict__ xc, const float* __restrict__ xo,   // [B,H] slices
    const float* __restrict__ Whi, const float* __restrict__ Whf,
    const float* __restrict__ Whc, const float* __restrict__ Who)
{
    const int lane = threadIdx.x & 31;
    const int mi = lane & 15;
    const int hi = lane >> 4;
    const int m0 = blockIdx.x * 16;
    const int n0 = blockIdx.y * 16;
    const int n  = n0 + mi;

    const float* ap = hprev + (size_t)(m0 + mi) * HIDDEN + 2 * hi;
    const size_t woff = (size_t)n * HIDDEN + 2 * hi;
    const float* bpi = Whi + woff;
    const float* bpf = Whf + woff;
    const float* bpc = Whc + woff;
    const float* bpo = Who + woff;

    v8f ai = {}, af = {}, ac = {}, ao = {};
    #pragma unroll 2
    for (int k = 0; k < HIDDEN; k += 4) {
        v2f a  = *(const v2f*)(ap + k);
        v2f bi = *(const v2f*)(bpi + k);
        v2f bf = *(const v2f*)(bpf + k);
        v2f bc = *(const v2f*)(bpc + k);
        v2f bo = *(const v2f*)(bpo + k);
        ai = wmma_f32(a, bi, ai);
        af = wmma_f32(a, bf, af);
        ac = wmma_f32(a, bc, ac);
        ao = wmma_f32(a, bo, ao);
    }

    #pragma unroll
    for (int j = 0; j < 8; ++j) {
        const int brow = m0 + j + 8 * hi;
        const size_t idx = (size_t)brow * HIDDEN + n;
        const float iv = sigm(xi[idx] + ai[j]);
        const float fv = sigm(xf[idx] + af[j]);
        const float ov = sigm(xo[idx] + ao[j]);
        const float gv = ftanh(xc[idx] + ac[j]);
        const float cn = fv * cbuf[idx] + iv * gv;
        cbuf[idx] = cn;
        hout[idx] = ov * ftanh(cn);
    }
}

// ---------------------------------------------------------------------------
// Kernel 3: logits = hs . Why^T + Why_b.  [8192 x 1024] x [1024 x 10000].
// Output row r = b*SEQ + s -> hs element (s = r&255, b = r>>8).
// One wave per block; 64(M) x 16(N) macro-tile = 4 accumulators sharing one
// Why fragment per k-step.  Grid = (8192/64, 10000/16).
// ---------------------------------------------------------------------------
__global__ __launch_bounds__(32) void logits_kernel(
    const float* __restrict__ hs, const float* __restrict__ Why,
    const float* __restrict__ Why_b, float* __restrict__ out)
{
    const int lane = threadIdx.x & 31;
    const int mi = lane & 15;
    const int hi = lane >> 4;
    const int m0 = blockIdx.x * 64;
    const int n0 = blockIdx.y * 16;

    const int n = n0 + mi;
    const float* bp = Why + (size_t)n * HIDDEN + 2 * hi;

    const float* ap[4];
    #pragma unroll
    for (int t = 0; t < 4; ++t) {
        const int r = m0 + 16 * t + mi;              // r = b*SEQ + s
        ap[t] = hs + (size_t)(r & 255) * (BATCH * HIDDEN)
                   + (size_t)(r >> 8) * HIDDEN + 2 * hi;
    }

    v8f acc0 = {}, acc1 = {}, acc2 = {}, acc3 = {};
    for (int k = 0; k < HIDDEN; k += 4) {
        if ((k & 255) == 0) {
            __builtin_prefetch(bp + k + 256, 0, 1);   // global_prefetch_b8
            __builtin_prefetch(ap[0] + k + 256, 0, 1);
        }
        v2f bf = *(const v2f*)(bp + k);
        v2f a0 = *(const v2f*)(ap[0] + k);
        v2f a1 = *(const v2f*)(ap[1] + k);
        v2f a2 = *(const v2f*)(ap[2] + k);
        v2f a3 = *(const v2f*)(ap[3] + k);
        acc0 = wmma_f32(a0, bf, acc0);
        acc1 = wmma_f32(a1, bf, acc1);
        acc2 = wmma_f32(a2, bf, acc2);
        acc3 = wmma_f32(a3, bf, acc3);
    }

    const float bias = Why_b[n];
    v8f accs[4] = {acc0, acc1, acc2, acc3};
    #pragma unroll
    for (int t = 0; t < 4; ++t) {
        #pragma unroll
        for (int j = 0; j < 8; ++j) {
            const int orow = m0 + 16 * t + j + 8 * hi;
            out[(size_t)orow * VOCAB + n] = accs[t][j] + bias;
        }
    }
}

// ---------------------------------------------------------------------------
extern "C" void kernel_launch(void* const* d_in, const int* in_sizes, int n_in,
                              void* d_out, int out_size, void* d_ws, size_t ws_size,
                              hipStream_t stream) {
    const int*   x    = (const int*)  d_in[0];
    const float* h0   = (const float*)d_in[1];
    const float* c0   = (const float*)d_in[2];
    const float* emb  = (const float*)d_in[3];
    const float* Wxi  = (const float*)d_in[4];
    const float* bxi  = (const float*)d_in[5];
    const float* Whi  = (const float*)d_in[6];
    const float* Wxf  = (const float*)d_in[7];
    const float* bxf  = (const float*)d_in[8];
    const float* Whf  = (const float*)d_in[9];
    const float* Wxc  = (const float*)d_in[10];
    const float* bxc  = (const float*)d_in[11];
    const float* Whc  = (const float*)d_in[12];
    const float* Wxo  = (const float*)d_in[13];
    const float* bxo  = (const float*)d_in[14];
    const float* Who  = (const float*)d_in[15];
    const float* Why  = (const float*)d_in[16];
    const float* Whyb = (const float*)d_in[17];

    float* out = (float*)d_out;
    float* ws  = (float*)d_ws;

    const size_t SBH = (size_t)SEQ * BATCH * HIDDEN;   // 8,388,608 floats
    float* xi   = ws;
    float* xf   = xi + SBH;
    float* xc   = xf + SBH;
    float* xo   = xc + SBH;
    float* hs   = xo + SBH;
    float* cbuf = hs + SBH;
    (void)ws_size; (void)in_sizes; (void)n_in; (void)out_size;

    // init running cell state from input (deterministic per call)
    hipMemcpyAsync(cbuf, c0, (size_t)BATCH * HIDDEN * sizeof(float),
                   hipMemcpyDeviceToDevice, stream);

    // 1) input-side projections for all 4 gates, all timesteps
    GateParams p;
    p.W[0] = Wxi; p.W[1] = Wxf; p.W[2] = Wxc; p.W[3] = Wxo;
    p.bias[0] = bxi; p.bias[1] = bxf; p.bias[2] = bxc; p.bias[3] = bxo;
    p.out[0] = xi; p.out[1] = xf; p.out[2] = xc; p.out[3] = xo;
    {
        dim3 grid((BATCH * SEQ) / 64, HIDDEN / 16, 4);
        gates_x_kernel<<<grid, 32, 0, stream>>>(x, emb, p);
    }

    // 2) sequential recurrence, one launch per step
    const size_t BH = (size_t)BATCH * HIDDEN;
    for (int s = 0; s < SEQ; ++s) {
        const float* hprev = (s == 0) ? h0 : (hs + (size_t)(s - 1) * BH);
        dim3 grid(BATCH / 16, HIDDEN / 16);
        lstm_step_kernel<<<grid, 32, 0, stream>>>(
            hprev, hs + (size_t)s * BH, cbuf,
            xi + (size_t)s * BH, xf + (size_t)s * BH,
            xc + (size_t)s * BH, xo + (size_t)s * BH,
            Whi, Whf, Whc, Who);
    }

    // 3) output logits
    {
        dim3 grid((BATCH * SEQ) / 64, VOCAB / 16);
        logits_kernel<<<grid, 32, 0, stream>>>(hs, Why, Whyb, out);
    }

    // 4) h_final = hs[S-1], c_final = cbuf, appended after logits
    float* hfin = out + (size_t)BATCH * SEQ * VOCAB;
    float* cfin = hfin + BH;
    hipMemcpyAsync(hfin, hs + (size_t)(SEQ - 1) * BH, BH * sizeof(float),
                   hipMemcpyDeviceToDevice, stream);
    hipMemcpyAsync(cfin, cbuf, BH * sizeof(float),
                   hipMemcpyDeviceToDevice, stream);
}